// RoutingCapsMultiInput_51754355916820
// MI455X (gfx1250) — compile-verified
//
#include <hip/hip_runtime.h>
#include <hip/hip_bf16.h>
#include <math.h>

// Shapes: x0,x1:[16,512,8] f32; W0,W1:[16,512,8,64] f32; bias:[16,1024,1024] f32
// out:[16,16,64] f32.
// Key identity: rowsum(softmax(A)) == 1, so S[b,k] = sum_i (1+bsum[k,i]) * (X_i @ W[k,i]).

typedef float v2f __attribute__((ext_vector_type(2)));
typedef float v8f __attribute__((ext_vector_type(8)));

#define N1     16
#define D1     64
#define NIN    1024   // total input capsules
#define HALFIN 512
#define BATCH  16

// ---------------------------------------------------------------------------
// Kernel 0: zero the S accumulation buffer (16*16*64 floats)
// ---------------------------------------------------------------------------
__global__ void zero_s(float* __restrict__ S) {
    S[blockIdx.x * 256 + threadIdx.x] = 0.0f;
}

// ---------------------------------------------------------------------------
// Kernel 1: w[k,i] = 1 + sum_z bias[k,i,z].  One wave per (k,i) row.
// blockDim = 256 (8 waves), grid = 16384/8 = 2048.
// ---------------------------------------------------------------------------
__global__ void bias_rowsum(const float* __restrict__ bias, float* __restrict__ wki) {
    const int lane = threadIdx.x & 31;
    const int wv   = threadIdx.x >> 5;
    const int row  = blockIdx.x * 8 + wv;        // 0 .. 16383  (= k*1024 + i)
    const float4* r4 = (const float4*)(bias + (size_t)row * 1024);
    float s = 0.0f;
#pragma unroll
    for (int q = 0; q < 8; ++q) {
        float4 v = r4[lane + 32 * q];
        s += v.x + v.y + v.z + v.w;
    }
#pragma unroll
    for (int off = 16; off > 0; off >>= 1)
        s += __shfl_xor(s, off, 32);
    if (lane == 0) wki[row] = 1.0f + s;
}

// ---------------------------------------------------------------------------
// Kernel 2: S[k][b][l] += sum_i (w[k,i] * X_i) @ W[k,i]  via V_WMMA_F32_16X16X4_F32.
// grid = (64 i-chunks, 16 k), blockDim = 256 (8 waves). Each wave owns 2 i's.
// Accumulators (4 N-tiles of 16x16 f32) live in VGPRs across the i loop;
// waves combine through LDS ds_add_f32, block commits via global atomic_add_f32.
// ---------------------------------------------------------------------------
__global__ void __launch_bounds__(256)
caps_fused_wmma(const float* __restrict__ x0, const float* __restrict__ x1,
                const float* __restrict__ W0, const float* __restrict__ W1,
                const float* __restrict__ wki, float* __restrict__ Sbuf) {
    const int lane = threadIdx.x & 31;
    const int wv   = threadIdx.x >> 5;           // 0..7
    const int m    = lane & 15;                  // A: batch row / B: column-in-tile
    const int hi   = lane >> 4;                  // lane half select
    const int koff = hi << 1;                    // 0 or 2 (K rows held by this lane)
    const int k    = blockIdx.y;                 // output capsule
    const int ibase = blockIdx.x * 16 + wv * 2;  // this wave's first i

    __shared__ float Stile[BATCH * D1];          // 16x64 partial S for this block
#pragma unroll
    for (int q = 0; q < 4; ++q)
        Stile[threadIdx.x + 256 * q] = 0.0f;
    __syncthreads();

    v8f acc0 = {}, acc1 = {}, acc2 = {}, acc3 = {};

#pragma unroll
    for (int ii = 0; ii < 2; ++ii) {
        const int i = ibase + ii;                // wave-uniform
        const float* xp;
        const float* Wp;
        if (i < HALFIN) {
            xp = x0 + ((size_t)m * HALFIN + i) * 8;
            Wp = W0 + (((size_t)k * HALFIN + i) * 8) * 64;
        } else {
            const int i2 = i - HALFIN;
            xp = x1 + ((size_t)m * HALFIN + i2) * 8;
            Wp = W1 + (((size_t)k * HALFIN + i2) * 8) * 64;
        }
        const float wv_ki = wki[k * NIN + i];

        // A operand: rows of (w * X_i), 16x4 f32 per K-half.
        v2f a_lo = { xp[koff]     * wv_ki, xp[koff + 1] * wv_ki };  // K = 0..3 half
        v2f a_hi = { xp[4 + koff] * wv_ki, xp[5 + koff] * wv_ki };  // K = 4..7 half

        // B operands per N-tile: rows koff/koff+1 of W[k,i] at column 16n+m.
#pragma unroll
        for (int n = 0; n < 4; ++n) {
            const int c = 16 * n + m;
            v2f b_lo = { Wp[(koff)     * 64 + c], Wp[(koff + 1) * 64 + c] };
            v2f b_hi = { Wp[(4 + koff) * 64 + c], Wp[(5 + koff) * 64 + c] };
            v8f cin = (n == 0) ? acc0 : (n == 1) ? acc1 : (n == 2) ? acc2 : acc3;
            cin = __builtin_amdgcn_wmma_f32_16x16x4_f32(false, a_lo, false, b_lo,
                                                        (short)0, cin, false, false);
            cin = __builtin_amdgcn_wmma_f32_16x16x4_f32(false, a_hi, false, b_hi,
                                                        (short)0, cin, false, false);
            if      (n == 0) acc0 = cin;
            else if (n == 1) acc1 = cin;
            else if (n == 2) acc2 = cin;
            else             acc3 = cin;
        }
    }

    // D layout: reg p, this lane -> S[row = p + 8*hi][col = 16n + m]
#pragma unroll
    for (int p = 0; p < 8; ++p) {
        const int row = (p + hi * 8) * 64 + m;
        atomicAdd(&Stile[row + 16 * 0], acc0[p]);
        atomicAdd(&Stile[row + 16 * 1], acc1[p]);
        atomicAdd(&Stile[row + 16 * 2], acc2[p]);
        atomicAdd(&Stile[row + 16 * 3], acc3[p]);
    }
    __syncthreads();

    float* Sp = Sbuf + (size_t)k * (BATCH * D1);
#pragma unroll
    for (int q = 0; q < 4; ++q) {
        const int idx = threadIdx.x + 256 * q;
        atomicAdd(&Sp[idx], Stile[idx]);
    }
}

// ---------------------------------------------------------------------------
// Kernel 3: squash.  One wave per (b,k); Sbuf is [k][b][64], out is [b][k][64].
// ---------------------------------------------------------------------------
__global__ void squash_out(const float* __restrict__ Sbuf, float* __restrict__ out) {
    const int pair = blockIdx.x;                 // 0..255
    const int b    = pair >> 4;
    const int k    = pair & 15;
    const int lane = threadIdx.x;
    const float* Sp = Sbuf + ((size_t)k * BATCH + b) * D1;
    float s0 = Sp[lane];
    float s1 = Sp[lane + 32];
    float part = s0 * s0 + s1 * s1;
#pragma unroll
    for (int off = 16; off > 0; off >>= 1)
        part += __shfl_xor(part, off, 32);
    const float sq    = part;
    const float scale = (sq / (1.0f + sq)) / (sqrtf(sq) + 1e-5f);
    float* op = out + ((size_t)b * N1 + k) * D1;
    op[lane]      = s0 * scale;
    op[lane + 32] = s1 * scale;
}

// ---------------------------------------------------------------------------
extern "C" void kernel_launch(void* const* d_in, const int* in_sizes, int n_in,
                              void* d_out, int out_size, void* d_ws, size_t ws_size,
                              hipStream_t stream) {
    const float* x0   = (const float*)d_in[0];
    const float* x1   = (const float*)d_in[1];
    const float* W0   = (const float*)d_in[2];
    const float* W1   = (const float*)d_in[3];
    const float* bias = (const float*)d_in[4];
    float* out  = (float*)d_out;

    float* Sbuf = (float*)d_ws;                  // 16*16*64  = 16384 floats
    float* wki  = Sbuf + (N1 * BATCH * D1);      // 16*1024   = 16384 floats

    zero_s<<<dim3((N1 * BATCH * D1) / 256), dim3(256), 0, stream>>>(Sbuf);
    bias_rowsum<<<dim3((N1 * NIN) / 8), dim3(256), 0, stream>>>(bias, wki);
    caps_fused_wmma<<<dim3(64, N1), dim3(256), 0, stream>>>(x0, x1, W0, W1, wki, Sbuf);
    squash_out<<<dim3(BATCH * N1), dim3(32), 0, stream>>>(Sbuf, out);
}